// EntmaxNsect_78709570667198
// MI455X (gfx1250) — compile-verified
//
#include <hip/hip_runtime.h>
#include <cstdint>

#define DLEN   32000
#define V4LEN  8000          // float4s per row
#define BLOCKT 1024          // 32 wave32s
#define NSLOT  8             // ceil(8000/1024)
#define NFULL  7             // 7*1024 = 7168
#define NREM   832           // 8000-7168; multiple of 32 -> wave-uniform split
#define NITER  5

// ---- CDNA5 async global->LDS (ASYNCcnt-tracked, per CDNA5 ISA) ----
__device__ __forceinline__ void async_ld_b128(unsigned lds_byte_off, const float4* gp) {
  asm volatile("global_load_async_to_lds_b128 %0, %1, off"
               :: "v"(lds_byte_off), "v"(gp)
               : "memory");
}

__device__ __forceinline__ float wave_max(float v) {
#pragma unroll
  for (int off = 16; off > 0; off >>= 1)
    v = fmaxf(v, __shfl_xor(v, off, 32));
  return v;
}

__device__ __forceinline__ float wave_sum(float v) {
#pragma unroll
  for (int off = 16; off > 0; off >>= 1)
    v += __shfl_xor(v, off, 32);
  return v;
}

#define PROBE1(xv)                                                  \
  do {                                                              \
    float d_;                                                       \
    d_ = (xv) - t1; d_ = fmaxf(d_, 0.0f); a1 = fmaf(d_, d_, a1);    \
    d_ = (xv) - t2; d_ = fmaxf(d_, 0.0f); a2 = fmaf(d_, d_, a2);    \
    d_ = (xv) - t3; d_ = fmaxf(d_, 0.0f); a3 = fmaf(d_, d_, a3);    \
    d_ = (xv) - t4; d_ = fmaxf(d_, 0.0f); a4 = fmaf(d_, d_, a4);    \
  } while (0)

__global__ __launch_bounds__(BLOCKT) void entmax15_nsect_kernel(
    const float* __restrict__ X, float* __restrict__ P) {
  __shared__ float4 stage[2][BLOCKT];   // 32 KB async staging double buffer
  __shared__ float4 red4[32];
  __shared__ float  red1[32];

  const int tid  = threadIdx.x;
  const int wid  = tid >> 5;
  const int lane = tid & 31;
  const size_t rowBase = (size_t)blockIdx.x * (size_t)V4LEN;
  const float4* __restrict__ xin  = reinterpret_cast<const float4*>(X) + rowBase;
  float4* __restrict__       pout = reinterpret_cast<float4*>(P) + rowBase;

  const int nslot = (tid < NREM) ? NSLOT : NFULL;  // wave-uniform

  const unsigned lds0 = (unsigned)(uintptr_t)&stage[0][tid];
  const unsigned lds1 = (unsigned)(uintptr_t)&stage[1][tid];

  // ---- depth-2 async pipeline: HBM -> LDS (async engine) -> VGPR ----
  async_ld_b128(lds0, xin + tid);
  async_ld_b128(lds1, xin + tid + BLOCKT);

  float4 xs[NSLOT];
  float mx = -3.0e38f;
#pragma unroll
  for (int k = 0; k < NSLOT; ++k) {
    if (k < nslot) {
      if (k == nslot - 1)
        asm volatile("s_wait_asynccnt 0" ::: "memory");  // last slot in flight
      else
        asm volatile("s_wait_asynccnt 1" ::: "memory");  // slot k complete (in-order)
      float4 v = stage[k & 1][tid];                      // ds_load_b128
      v.x *= 0.5f; v.y *= 0.5f; v.z *= 0.5f; v.w *= 0.5f;  // Xs = (alpha-1)*X
      xs[k] = v;
      mx = fmaxf(mx, fmaxf(fmaxf(v.x, v.y), fmaxf(v.z, v.w)));
      // async LDS writes are unordered vs DS reads: make sure the ds_load fully
      // retired before the buffer half is overwritten by the next async load.
      asm volatile("s_wait_dscnt 0" ::: "memory");
      if (k + 2 < nslot)
        async_ld_b128((k & 1) ? lds1 : lds0, xin + tid + (k + 2) * BLOCKT);
    } else {
      xs[k] = make_float4(-1.0e30f, -1.0e30f, -1.0e30f, -1.0e30f);  // contributes 0
    }
  }

  // ---- block max ----
  mx = wave_max(mx);
  if (lane == 0) red1[wid] = mx;
  __syncthreads();
  if (tid < 32) {
    float u = wave_max(red1[tid]);
    if (tid == 0) red1[0] = u;
  }
  __syncthreads();
  const float mrow = red1[0];

  float tau_lo = mrow - 1.0f;
  float tau_hi = mrow - 0.005590169943749474f;  // (1/32000)^(alpha-1)

  // ---- 5 iterations of 5-section bisection; 4 probes fused into one reg pass ----
#pragma unroll 1
  for (int it = 0; it < NITER; ++it) {
    const float width = (tau_hi - tau_lo) / 5.0f;
    const float t1 = tau_lo + width;
    const float t2 = tau_lo + 2.0f * width;
    const float t3 = tau_lo + 3.0f * width;
    const float t4 = tau_lo + 4.0f * width;
    float a1 = 0.0f, a2 = 0.0f, a3 = 0.0f, a4 = 0.0f;
#pragma unroll
    for (int k = 0; k < NSLOT; ++k) {
      const float4 v = xs[k];
      PROBE1(v.x); PROBE1(v.y); PROBE1(v.z); PROBE1(v.w);
    }
    a1 = wave_sum(a1); a2 = wave_sum(a2); a3 = wave_sum(a3); a4 = wave_sum(a4);
    __syncthreads();  // red4 free from previous iteration's broadcast
    if (lane == 0) red4[wid] = make_float4(a1, a2, a3, a4);
    __syncthreads();
    if (tid < 32) {
      float4 u = red4[tid];
      u.x = wave_sum(u.x); u.y = wave_sum(u.y);
      u.z = wave_sum(u.z); u.w = wave_sum(u.w);
      if (tid == 0) red4[0] = u;
    }
    __syncthreads();
    const float4 m = red4[0];
    // j_best = largest j with mass(tau_lo + j*width) >= 1 (reference semantics)
    float jb = 0.0f;
    if (m.x >= 1.0f) jb = 1.0f;
    if (m.y >= 1.0f) jb = 2.0f;
    if (m.z >= 1.0f) jb = 3.0f;
    if (m.w >= 1.0f) jb = 4.0f;
    tau_lo = tau_lo + jb * width;
    tau_hi = tau_lo + width;
  }

  // ---- final p = relu(Xs - tau_lo)^2, normalized ----
  float ssum = 0.0f;
#pragma unroll
  for (int k = 0; k < NSLOT; ++k) {
    const float4 v = xs[k];
    float4 p; float d;
    d = v.x - tau_lo; d = fmaxf(d, 0.0f); p.x = d * d; ssum += p.x;
    d = v.y - tau_lo; d = fmaxf(d, 0.0f); p.y = d * d; ssum += p.y;
    d = v.z - tau_lo; d = fmaxf(d, 0.0f); p.z = d * d; ssum += p.z;
    d = v.w - tau_lo; d = fmaxf(d, 0.0f); p.w = d * d; ssum += p.w;
    xs[k] = p;  // reuse registers
  }
  ssum = wave_sum(ssum);
  __syncthreads();
  if (lane == 0) red1[wid] = ssum;
  __syncthreads();
  if (tid < 32) {
    float u = wave_sum(red1[tid]);
    if (tid == 0) red1[0] = u;
  }
  __syncthreads();
  const float inv = 1.0f / red1[0];

#pragma unroll
  for (int k = 0; k < NSLOT; ++k) {
    if (k < nslot) {
      float4 p = xs[k];
      p.x *= inv; p.y *= inv; p.z *= inv; p.w *= inv;
      pout[tid + k * BLOCKT] = p;  // global_store_b128
    }
  }
}

extern "C" void kernel_launch(void* const* d_in, const int* in_sizes, int n_in,
                              void* d_out, int out_size, void* d_ws, size_t ws_size,
                              hipStream_t stream) {
  (void)n_in; (void)out_size; (void)d_ws; (void)ws_size;
  const float* X = (const float*)d_in[0];
  float* P = (float*)d_out;
  const int rows = in_sizes[0] / DLEN;  // 4*2048 = 8192
  entmax15_nsect_kernel<<<rows, BLOCKT, 0, stream>>>(X, P);
}